// Head_19619410608768
// MI455X (gfx1250) — compile-verified
//
#include <hip/hip_runtime.h>
#include <math.h>

#define B_DIM 256
#define N_DIM 8192
#define M_DIM 64
#define H_DIM 512

typedef __attribute__((ext_vector_type(2))) float v2f;
typedef __attribute__((ext_vector_type(4))) float v4f;
typedef __attribute__((ext_vector_type(8))) float v8f;

// ---------------------------------------------------------------------------
// Kernel 1: k = x @ Wk + bk via V_WMMA_F32_16X16X4_F32.
// Tiles: 16 M-tiles (batch) x 4 N-tiles (cols), 1 wave each, K-loop H/4=128.
// A (16x4 f32): lanes 0-15 hold K=kk+0,kk+1; lanes 16-31 hold K=kk+2,kk+3.
// B (4x16 f32): mirrored: lane half selects K pair, vgpr selects K within pair.
// C/D (16x16 f32): VGPR v -> M = v + 8*(lane/16), N = lane%16.
// ---------------------------------------------------------------------------
__global__ __launch_bounds__(32)
void k_gemm_wmma(const float* __restrict__ x, const float* __restrict__ Wk,
                 const float* __restrict__ bk, float* __restrict__ kout) {
    const int tileM = blockIdx.x & 15;
    const int tileN = blockIdx.x >> 4;
    const int lane  = threadIdx.x;
    const int half  = lane >> 4;
    const int l16   = lane & 15;

    const int rowA = tileM * 16 + l16;
    const int colB = tileN * 16 + l16;
    const float* xrow = x + rowA * H_DIM;

    v8f c = {0.f, 0.f, 0.f, 0.f, 0.f, 0.f, 0.f, 0.f};
    for (int kk = 0; kk < H_DIM; kk += 4) {
        const int kbase = kk + 2 * half;
        v2f a;
        a.x = xrow[kbase + 0];
        a.y = xrow[kbase + 1];
        v2f bm;
        bm.x = Wk[(kbase + 0) * M_DIM + colB];
        bm.y = Wk[(kbase + 1) * M_DIM + colB];
        c = __builtin_amdgcn_wmma_f32_16x16x4_f32(false, a, false, bm,
                                                  (short)0, c, false, false);
    }
    const float bias = bk[colB];
#pragma unroll
    for (int v = 0; v < 8; ++v) {
        const int m = tileM * 16 + v + 8 * half;
        kout[m * M_DIM + colB] = c[v] + bias;
    }
}

// ---------------------------------------------------------------------------
// Kernel 2: per-batch scalar projections beta/g/gamma/s and nb = ||k+eps||.
// One 128-thread block per batch row. 7 simultaneous reductions.
// ---------------------------------------------------------------------------
__device__ __forceinline__ float softplus_f(float v) {
    return fmaxf(v, 0.f) + log1pf(expf(-fabsf(v)));
}

__global__ __launch_bounds__(128)
void proj_kernel(const float* __restrict__ x, const float* __restrict__ kmat,
                 const float* __restrict__ Wbeta, const float* __restrict__ bbeta,
                 const float* __restrict__ Wg, const float* __restrict__ bg,
                 const float* __restrict__ Ws, const float* __restrict__ bs,
                 const float* __restrict__ Wgamma, const float* __restrict__ bgamma,
                 float* __restrict__ beta_o, float* __restrict__ g_o,
                 float* __restrict__ gamma_o, float* __restrict__ s_o,
                 float* __restrict__ nb_o) {
    const int b = blockIdx.x, t = threadIdx.x;
    const float* xb = x + b * H_DIM;

    float p0 = 0.f, p1 = 0.f, p2 = 0.f, p3 = 0.f, p4 = 0.f, p5 = 0.f, p6 = 0.f;
    for (int h = t; h < H_DIM; h += 128) {
        const float xv = xb[h];
        p0 += xv * Wbeta[h];
        p1 += xv * Wg[h];
        p2 += xv * Wgamma[h];
        p3 += xv * Ws[3 * h + 0];
        p4 += xv * Ws[3 * h + 1];
        p5 += xv * Ws[3 * h + 2];
    }
    if (t < M_DIM) {
        const float kv = kmat[b * M_DIM + t] + 1e-16f;
        p6 = kv * kv;
    }

    __shared__ float red[7][4];
    float pr[7] = {p0, p1, p2, p3, p4, p5, p6};
    const int wid = t >> 5, lane = t & 31;
#pragma unroll
    for (int j = 0; j < 7; ++j) {
        float v = pr[j];
#pragma unroll
        for (int off = 16; off >= 1; off >>= 1) v += __shfl_xor(v, off, 32);
        if (lane == 0) red[j][wid] = v;
    }
    __syncthreads();
    if (t == 0) {
        float s[7];
#pragma unroll
        for (int j = 0; j < 7; ++j)
            s[j] = red[j][0] + red[j][1] + red[j][2] + red[j][3];
        beta_o[b]  = softplus_f(s[0] + bbeta[0]);
        g_o[b]     = 1.f / (1.f + expf(-(s[1] + bg[0])));
        gamma_o[b] = 1.f + softplus_f(s[2] + bgamma[0]);
        const float a0 = s[3] + bs[0], a1 = s[4] + bs[1], a2 = s[5] + bs[2];
        const float mx = fmaxf(a0, fmaxf(a1, a2));
        const float e0 = expf(a0 - mx), e1 = expf(a1 - mx), e2 = expf(a2 - mx);
        const float inv = 1.f / (e0 + e1 + e2);
        s_o[3 * b + 0] = e0 * inv;
        s_o[3 * b + 1] = e1 * inv;
        s_o[3 * b + 2] = e2 * inv;
        nb_o[b] = sqrtf(s[6]);
    }
}

// ---------------------------------------------------------------------------
// Kernel 3: THE bandwidth kernel. Single fused pass over memory_read (512MB):
//   z[b,n] = beta[b] * dot((mem[b,n,:]+eps),(k[b,:]+eps))
//                    / (max(||mem row||,1e-8) * max(nb[b],1e-8))
// 8-lane groups: each lane owns 8 contiguous elems (two b128 NT loads), a
// wave covers 4 rows = 1KB contiguous per iteration. Reduction = 3 xor
// shuffles per value (masks 1,2,4 stay inside the 8-lane group), i.e. only
// 6 DS ops per 1KB of HBM traffic.
// ---------------------------------------------------------------------------
__global__ __launch_bounds__(256)
void content_kernel(const float* __restrict__ mem, const float* __restrict__ kmat,
                    const float* __restrict__ beta, const float* __restrict__ nb,
                    float* __restrict__ z) {
    const int b    = blockIdx.x;
    const int t    = threadIdx.x;
    const int wid  = t >> 5;
    const int lane = t & 31;
    const int grp  = lane >> 3;   // 4 groups of 8 lanes -> 4 rows per iter
    const int li   = lane & 7;    // lane within group: elems [8*li, 8*li+8)

    v4f kq0 = *(const v4f*)(kmat + b * M_DIM + 8 * li);
    v4f kq1 = *(const v4f*)(kmat + b * M_DIM + 8 * li + 4);
    kq0.x += 1e-16f; kq0.y += 1e-16f; kq0.z += 1e-16f; kq0.w += 1e-16f;
    kq1.x += 1e-16f; kq1.y += 1e-16f; kq1.z += 1e-16f; kq1.w += 1e-16f;

    const float scale = beta[b] / fmaxf(nb[b], 1e-8f);
    const float* memb = mem + (size_t)b * N_DIM * M_DIM;
    const int baseRow = blockIdx.y * 256 + wid * 32;   // 8 waves * 32 rows

#pragma unroll 4
    for (int it = 0; it < 8; ++it) {
        const int row = baseRow + it * 4 + grp;
        const float* rp = memb + (size_t)row * M_DIM + 8 * li;
        const v4f m0 = __builtin_nontemporal_load((const v4f*)rp);
        const v4f m1 = __builtin_nontemporal_load((const v4f*)(rp + 4));
        const float a0 = m0.x + 1e-16f, a1 = m0.y + 1e-16f;
        const float a2 = m0.z + 1e-16f, a3 = m0.w + 1e-16f;
        const float a4 = m1.x + 1e-16f, a5 = m1.y + 1e-16f;
        const float a6 = m1.z + 1e-16f, a7 = m1.w + 1e-16f;
        float dotp = a0 * kq0.x + a1 * kq0.y + a2 * kq0.z + a3 * kq0.w
                   + a4 * kq1.x + a5 * kq1.y + a6 * kq1.z + a7 * kq1.w;
        float nap  = a0 * a0 + a1 * a1 + a2 * a2 + a3 * a3
                   + a4 * a4 + a5 * a5 + a6 * a6 + a7 * a7;
#pragma unroll
        for (int off = 4; off >= 1; off >>= 1) {   // stays inside 8-lane group
            dotp += __shfl_xor(dotp, off, 32);
            nap  += __shfl_xor(nap,  off, 32);
        }
        if (li == 0) {
            const float na = sqrtf(nap);
            z[b * N_DIM + row] = scale * dotp / fmaxf(na, 1e-8f);
        }
    }
}

// ---------------------------------------------------------------------------
// Kernel 4: per-batch finalize: softmax over N, interpolate with prev,
// 3-tap circular shift (via 32KB LDS row), sharpen (pow gamma), normalize.
// One 256-thread block per batch, 32 elems/thread (interleaved, coalesced).
// ---------------------------------------------------------------------------
__device__ __forceinline__ float block_reduce_add(float v, float* red) {
#pragma unroll
    for (int off = 16; off >= 1; off >>= 1) v += __shfl_xor(v, off, 32);
    const int wid = threadIdx.x >> 5, lane = threadIdx.x & 31;
    if (lane == 0) red[wid] = v;
    __syncthreads();
    if (wid == 0) {
        float x = (lane < 8) ? red[lane] : 0.f;
#pragma unroll
        for (int off = 4; off >= 1; off >>= 1) x += __shfl_xor(x, off, 32);
        if (lane == 0) red[0] = x;
    }
    __syncthreads();
    const float r = red[0];
    __syncthreads();
    return r;
}

__device__ __forceinline__ float block_reduce_max(float v, float* red) {
#pragma unroll
    for (int off = 16; off >= 1; off >>= 1) v = fmaxf(v, __shfl_xor(v, off, 32));
    const int wid = threadIdx.x >> 5, lane = threadIdx.x & 31;
    if (lane == 0) red[wid] = v;
    __syncthreads();
    if (wid == 0) {
        float x = (lane < 8) ? red[lane] : -3.402823466e38f;
#pragma unroll
        for (int off = 4; off >= 1; off >>= 1) x = fmaxf(x, __shfl_xor(x, off, 32));
        if (lane == 0) red[0] = x;
    }
    __syncthreads();
    const float r = red[0];
    __syncthreads();
    return r;
}

__global__ __launch_bounds__(256)
void finalize_kernel(const float* __restrict__ z, const float* __restrict__ prev,
                     const float* __restrict__ g_a, const float* __restrict__ s_a,
                     const float* __restrict__ gamma_a, float* __restrict__ out) {
    __shared__ float sW[N_DIM];   // 32 KB: w_g row for circular shift
    __shared__ float red[8];
    const int b = blockIdx.x, t = threadIdx.x;
    const float* zb = z + (size_t)b * N_DIM;
    const float* pb = prev + (size_t)b * N_DIM;
    const float g = g_a[b], gamma = gamma_a[b];
    const float s0 = s_a[3 * b], s1 = s_a[3 * b + 1], s2 = s_a[3 * b + 2];

    float zl[32];
    float m = -3.402823466e38f;
#pragma unroll
    for (int i = 0; i < 32; ++i) {
        zl[i] = zb[i * 256 + t];
        m = fmaxf(m, zl[i]);
    }
    m = block_reduce_max(m, red);

    float sum = 0.f;
#pragma unroll
    for (int i = 0; i < 32; ++i) {
        const float e = expf(zl[i] - m);
        sW[i * 256 + t] = e;
        sum += e;
    }
    sum = block_reduce_add(sum, red);   // barrier => all sW writes visible
    const float invs = 1.f / sum;

#pragma unroll
    for (int i = 0; i < 32; ++i) {
        const int n = i * 256 + t;
        sW[n] = g * sW[n] * invs + (1.f - g) * pb[n];   // w_g
    }
    __syncthreads();

    float pv[32];
    float sum2 = 0.f;
#pragma unroll
    for (int i = 0; i < 32; ++i) {
        const int n  = i * 256 + t;
        const int nm = (n == 0) ? (N_DIM - 1) : n - 1;
        const int np = (n == N_DIM - 1) ? 0 : n + 1;
        const float wt = s0 * sW[nm] + s1 * sW[n] + s2 * sW[np];
        const float p  = powf(wt, gamma);
        pv[i] = p;
        sum2 += p;
    }
    sum2 = block_reduce_add(sum2, red);
    const float invn = 1.f / (sum2 + 1e-16f);
#pragma unroll
    for (int i = 0; i < 32; ++i)
        out[(size_t)b * N_DIM + i * 256 + t] = pv[i] * invn;
}

// ---------------------------------------------------------------------------
extern "C" void kernel_launch(void* const* d_in, const int* in_sizes, int n_in,
                              void* d_out, int out_size, void* d_ws, size_t ws_size,
                              hipStream_t stream) {
    const float* x      = (const float*)d_in[0];
    const float* prev   = (const float*)d_in[1];
    const float* mem    = (const float*)d_in[2];
    const float* Wk     = (const float*)d_in[3];
    const float* bk     = (const float*)d_in[4];
    const float* Wbeta  = (const float*)d_in[5];
    const float* bbeta  = (const float*)d_in[6];
    const float* Wg     = (const float*)d_in[7];
    const float* bg     = (const float*)d_in[8];
    const float* Ws     = (const float*)d_in[9];
    const float* bs     = (const float*)d_in[10];
    const float* Wgamma = (const float*)d_in[11];
    const float* bgamma = (const float*)d_in[12];
    float* out = (float*)d_out;

    float* ws     = (float*)d_ws;
    float* k_ws   = ws;            ws += B_DIM * M_DIM;   // 16384
    float* beta_w = ws;            ws += B_DIM;
    float* g_w    = ws;            ws += B_DIM;
    float* gamma_w= ws;            ws += B_DIM;
    float* nb_w   = ws;            ws += B_DIM;
    float* s_w    = ws;            ws += 3 * B_DIM;
    float* z_w    = ws;                                    // B*N = 2M floats

    k_gemm_wmma<<<64, 32, 0, stream>>>(x, Wk, bk, k_ws);
    proj_kernel<<<B_DIM, 128, 0, stream>>>(x, k_ws, Wbeta, bbeta, Wg, bg,
                                           Ws, bs, Wgamma, bgamma,
                                           beta_w, g_w, gamma_w, s_w, nb_w);
    content_kernel<<<dim3(B_DIM, 32), 256, 0, stream>>>(mem, k_ws, beta_w, nb_w, z_w);
    finalize_kernel<<<B_DIM, 256, 0, stream>>>(z_w, prev, g_w, s_w, gamma_w, out);
}